// LengthRegulator_56925496541440
// MI455X (gfx1250) — compile-verified
//
#include <hip/hip_runtime.h>

// LengthRegulator for MI455X (gfx1250).
// Pure data-mover: ~200MB traffic @ 23.3 TB/s => ~8us floor. Strategy:
// stage each gathered 1536B row through LDS with the CDNA5 async
// global<->LDS DMA path (ASYNCcnt-tracked), no VGPR data movement.

#define B_ 16
#define T_IN_ 2048
#define D_ 384          // floats per row => 1536 bytes
#define OUT_LEN_ 4096
#define ROW_BYTES_ 1536

#if defined(__has_builtin)
# if __has_builtin(__builtin_amdgcn_global_load_async_to_lds_b128) && \
     __has_builtin(__builtin_amdgcn_global_store_async_from_lds_b128)
#  define HAVE_ASYNC_LDS 1
# endif
# if __has_builtin(__builtin_amdgcn_s_wait_asynccnt)
#  define HAVE_WAIT_ASYNC 1
# endif
#endif

typedef int v4i_ __attribute__((vector_size(16)));
typedef __attribute__((address_space(1))) v4i_* gvec_p;   // global v4i*
typedef __attribute__((address_space(3))) v4i_* lvec_p;   // LDS v4i*

__device__ __forceinline__ void wait_asynccnt0() {
#if defined(HAVE_WAIT_ASYNC)
  __builtin_amdgcn_s_wait_asynccnt(0);
#else
  asm volatile("s_wait_asynccnt 0" ::: "memory");
#endif
}

// ---------------- Kernel 1: per-batch inclusive cumsum (16 x 2048) ----------
__global__ void lr_cumsum_kernel(const int* __restrict__ rc,
                                 int* __restrict__ cum,
                                 float* __restrict__ seq_out) {
  const int b   = blockIdx.x;      // 16 blocks
  const int tid = threadIdx.x;     // 256 threads, 8 elems each
  __shared__ int part[256];

  const int base = b * T_IN_ + tid * 8;
  int v[8];
  int run = 0;
#pragma unroll
  for (int i = 0; i < 8; ++i) { run += rc[base + i]; v[i] = run; }
  part[tid] = run;
  __syncthreads();

  // Hillis-Steele inclusive scan over 256 thread totals
  for (int off = 1; off < 256; off <<= 1) {
    int t = (tid >= off) ? part[tid - off] : 0;
    __syncthreads();
    part[tid] += t;
    __syncthreads();
  }
  const int excl = (tid == 0) ? 0 : part[tid - 1];
#pragma unroll
  for (int i = 0; i < 8; ++i) cum[base + i] = v[i] + excl;

  if (tid == 255) seq_out[b] = (float)part[255];  // output_seq_len (2nd output)
}

// ---------------- Kernel 2: binary search + async row gather ---------------
// grid = (OUT_LEN/8, B), block = 256 (8 wave32s); one output row per wave.
__global__ void lr_gather_kernel(const float* __restrict__ x,
                                 const int* __restrict__ cum,
                                 float* __restrict__ out) {
  const int b = blockIdx.y;
  __shared__ int    scum[T_IN_];                 // 8 KB, batch cumsum
  __shared__ float4 rowbuf[8][ROW_BYTES_ / 16];  // 12 KB, one 1536B row per wave

  // Cooperative stage of this batch's cumsum into LDS (L2-resident source).
  for (int i = threadIdx.x; i < T_IN_; i += 256) scum[i] = cum[b * T_IN_ + i];
  __syncthreads();

  const int wave = threadIdx.x >> 5;
  const int lane = threadIdx.x & 31;
  const int pos  = blockIdx.x * 8 + wave;        // wave-uniform
  const int seqlen = scum[T_IN_ - 1];

  // searchsorted(cum, pos, side='right'): first i with cum[i] > pos.
  int lo = 0, hi = T_IN_;
  while (lo < hi) {
    int mid = (lo + hi) >> 1;
    if (scum[mid] <= pos) lo = mid + 1; else hi = mid;
  }
  const int idx = (lo < T_IN_ - 1) ? lo : (T_IN_ - 1);

  float* dst = out + ((size_t)(b * OUT_LEN_ + pos)) * D_;

  if (pos < seqlen) {                            // wave-uniform branch
    const float* src = x + ((size_t)(b * T_IN_ + idx)) * D_;
#if defined(HAVE_ASYNC_LDS)
    // CDNA5 async DMA: global -> LDS -> global, 512B per instruction
    // (32 lanes x b128). Byte IOFFSET applies to BOTH addresses.
    gvec_p gs = (gvec_p)src + lane;
    lvec_p lb = (lvec_p)&rowbuf[wave][0] + lane;
    __builtin_amdgcn_global_load_async_to_lds_b128(gs, lb, 0,    0);
    __builtin_amdgcn_global_load_async_to_lds_b128(gs, lb, 512,  0);
    __builtin_amdgcn_global_load_async_to_lds_b128(gs, lb, 1024, 0);
    wait_asynccnt0();                            // LDS writes visible
    gvec_p gd = (gvec_p)dst + lane;
    __builtin_amdgcn_global_store_async_from_lds_b128(gd, lb, 0,    0);
    __builtin_amdgcn_global_store_async_from_lds_b128(gd, lb, 512,  0);
    __builtin_amdgcn_global_store_async_from_lds_b128(gd, lb, 1024, 0);
    // S_ENDPGM performs an implicit wait-idle; stores drain before exit.
#else
    const float4* s4 = (const float4*)src;
    float4*       d4 = (float4*)dst;
    d4[lane]      = s4[lane];
    d4[lane + 32] = s4[lane + 32];
    d4[lane + 64] = s4[lane + 64];
#endif
  } else {
    const float4 z = make_float4(0.f, 0.f, 0.f, 0.f);
    float4* d4 = (float4*)dst;
    d4[lane]      = z;
    d4[lane + 32] = z;
    d4[lane + 64] = z;
  }
}

// ---------------------------------------------------------------------------
extern "C" void kernel_launch(void* const* d_in, const int* in_sizes, int n_in,
                              void* d_out, int out_size, void* d_ws, size_t ws_size,
                              hipStream_t stream) {
  const float* x  = (const float*)d_in[0];        // (16, 2048, 384) f32
  const int*   rc = (const int*)d_in[1];          // (16, 2048) i32
  (void)in_sizes; (void)n_in; (void)out_size; (void)ws_size;

  int*   cum     = (int*)d_ws;                    // 16*2048 ints = 128 KB scratch
  float* out     = (float*)d_out;                 // (16, 4096, 384) f32
  float* seq_out = out + (size_t)B_ * OUT_LEN_ * D_;  // 16 seq lens (tail)

  lr_cumsum_kernel<<<B_, 256, 0, stream>>>(rc, cum, seq_out);

  dim3 grid(OUT_LEN_ / 8, B_);
  lr_gather_kernel<<<grid, 256, 0, stream>>>(x, cum, out);
}